// HistogramLoss_42185168781835
// MI455X (gfx1250) — compile-verified
//
#include <hip/hip_runtime.h>
#include <hip/hip_bf16.h>

typedef __attribute__((ext_vector_type(2))) float v2f;
typedef __attribute__((ext_vector_type(8))) float v8f;

#define NB 64                       // bins
#define NBATCH 64
#define PIX_PER_BATCH (512 * 512)
#define BPB 32                      // blocks per batch
#define CHUNK (PIX_PER_BATCH / BPB) // 8192 pixels per block
#define HT 256                      // histogram block size (8 waves)

// ---------------------------------------------------------------------------
// Zero the global count buffers (2 tensors x 64 batches x 64 bins, u32).
// ---------------------------------------------------------------------------
__global__ void zero_counts_kernel(unsigned int* __restrict__ cnt) {
  int i = blockIdx.x * blockDim.x + threadIdx.x;
  if (i < 2 * NBATCH * NB) cnt[i] = 0u;
}

// Exact searchsorted(right)-1 semantics: returns i with se[i] <= v < se[i+1],
// clamped to [0,63]; v == se[64] lands in bin 63.  Affine guess is exact for
// uniform edges; fix-up loops make it correct for any monotone edges.
__device__ __forceinline__ int bin_of(float v, const float* __restrict__ se,
                                      float e0, float invw) {
  float f = (v - e0) * invw;
  f = fmaxf(0.0f, fminf(f, (float)(NB - 1)));
  int i = (int)f;
  while (i > 0 && v < se[i]) --i;
  while (i < NB - 1 && v >= se[i + 1]) ++i;
  return i;
}

__device__ __forceinline__ void acc_one(float v, unsigned char m,
                                        const float* __restrict__ se,
                                        float e0, float eL, float invw,
                                        unsigned int* __restrict__ hw) {
  if (m && v >= e0 && v <= eL) {
    atomicAdd(&hw[bin_of(v, se, e0, invw)], 1u);
  }
}

// ---------------------------------------------------------------------------
// Streaming histogram: per-wave privatized LDS histograms, integer atomics
// (deterministic), float4 loads for bandwidth. Memory-bound: ~150MB total.
// ---------------------------------------------------------------------------
__global__ void __launch_bounds__(HT)
hist_kernel(const float* __restrict__ obs, const float* __restrict__ pred,
            const unsigned char* __restrict__ mask,
            const float* __restrict__ edges, unsigned int* __restrict__ cnt) {
  __shared__ unsigned int h[2][HT / 32][NB]; // [tensor][wave][bin]
  __shared__ float se[NB + 1];

  const int tid = threadIdx.x;
  const int wave = tid >> 5;

  for (int i = tid; i < 2 * (HT / 32) * NB; i += HT)
    (&h[0][0][0])[i] = 0u;
  if (tid < NB + 1) se[tid] = edges[tid];
  __syncthreads();

  const int batch = blockIdx.x / BPB;
  const int chunk = blockIdx.x % BPB;
  const long base = (long)batch * PIX_PER_BATCH + (long)chunk * CHUNK;

  const float4* o4 = (const float4*)(obs + base);
  const float4* p4 = (const float4*)(pred + base);
  const uchar4* m4 = (const uchar4*)(mask + base);

  const float e0 = se[0];
  const float eL = se[NB];
  const float invw = (float)NB / (eL - e0);

  unsigned int* ho = h[0][wave];
  unsigned int* hp = h[1][wave];

#pragma unroll
  for (int it = 0; it < CHUNK / 4 / HT; ++it) { // 8 iterations
    int q = it * HT + tid;
    float4 vo = o4[q];
    float4 vp = p4[q];
    uchar4 mm = m4[q];
    acc_one(vo.x, mm.x, se, e0, eL, invw, ho);
    acc_one(vo.y, mm.y, se, e0, eL, invw, ho);
    acc_one(vo.z, mm.z, se, e0, eL, invw, ho);
    acc_one(vo.w, mm.w, se, e0, eL, invw, ho);
    acc_one(vp.x, mm.x, se, e0, eL, invw, hp);
    acc_one(vp.y, mm.y, se, e0, eL, invw, hp);
    acc_one(vp.z, mm.z, se, e0, eL, invw, hp);
    acc_one(vp.w, mm.w, se, e0, eL, invw, hp);
  }
  __syncthreads();

  // Reduce the 8 wave-private histograms, one global integer atomic per bin.
  if (tid < 2 * NB) {
    int t = tid < NB ? 0 : 1;
    int b = tid & (NB - 1);
    unsigned int s = 0;
    for (int w = 0; w < HT / 32; ++w) s += h[t][w][b];
    atomicAdd(&cnt[t * (NBATCH * NB) + batch * NB + b], s);
  }
}

// ---------------------------------------------------------------------------
// Finalize: proportions -> CDF (via f32 WMMA against triangular-ones B) ->
// weighted squared diff -> loss.  One block, 512 threads = 16 waves; each
// wave owns one 16x16 tile of the 64x64 CDF matrices (obs & pred together).
// ---------------------------------------------------------------------------
__global__ void __launch_bounds__(512)
finalize_kernel(const unsigned int* __restrict__ cnt,
                const int* __restrict__ hidx_p,
                const float* __restrict__ mids,
                const float* __restrict__ bweights,
                float* __restrict__ out) {
  __shared__ float sP[2][NBATCH][NB + 1]; // +1 pad: kill LDS bank conflicts
  __shared__ float spart[512];

  const int tid = threadIdx.x;

  // ---- Step 1: per-batch normalization with empty-sample fallback ----------
  if (tid < 2 * NBATCH) {
    int t = tid >> 6;  // 0 = obs, 1 = pred
    int b = tid & (NBATCH - 1);
    const unsigned int* c = cnt + t * (NBATCH * NB) + b * NB;
    unsigned int tot = 0;
    for (int j = 0; j < NB; ++j) tot += c[j];
    float* dst = out + 1 + t * (NBATCH * NB) + b * NB;
    if (tot == 0) { // counts := all-ones, total := 64
      for (int j = 0; j < NB; ++j) { sP[t][b][j] = 1.0f / NB; dst[j] = 1.0f / NB; }
    } else {
      float inv = 1.0f / (float)tot;
      for (int j = 0; j < NB; ++j) {
        float p = (float)c[j] * inv;
        sP[t][b][j] = p;
        dst[j] = p;
      }
    }
  }
  __syncthreads();

  // ---- Step 2: CDF = P * T  (T[i][j] = i<=j) via V_WMMA_F32_16X16X4_F32 ----
  const int wave = tid >> 5;
  const int lane = tid & 31;
  const int mt = wave >> 2;        // output tile row   (batch / 16)
  const int nt = wave & 3;         // output tile col   (bin   / 16)
  const int mn = lane & 15;        // M (for A) / N (for B,D) within tile
  const int koff = (lane >> 4) * 2; // lanes 16-31 hold K=2,3 per ISA layout

  const int row = mt * 16 + mn;    // A-matrix row (batch)
  const int col = nt * 16 + mn;    // B/D column (bin)

  v8f co = {};
  v8f cp = {};
#pragma unroll
  for (int kk = 0; kk < 16; ++kk) {
    const int k0 = kk * 4 + koff;
    v2f ao, ap, bb;
    ao.x = sP[0][row][k0];
    ao.y = sP[0][row][k0 + 1];
    ap.x = sP[1][row][k0];
    ap.y = sP[1][row][k0 + 1];
    bb.x = (k0 <= col) ? 1.0f : 0.0f;       // triangular ones, exact in f32
    bb.y = (k0 + 1 <= col) ? 1.0f : 0.0f;
    co = __builtin_amdgcn_wmma_f32_16x16x4_f32(false, ao, false, bb,
                                               (short)0, co, false, false);
    cp = __builtin_amdgcn_wmma_f32_16x16x4_f32(false, ap, false, bb,
                                               (short)0, cp, false, false);
  }

  // ---- Step 3: weighted squared CDF difference -----------------------------
  const int h = *hidx_p;
  const float bw = (col < NB - 1) ? (mids[col + 1] - mids[col])
                                  : (mids[NB - 1] - mids[NB - 2]);
  const float wt = bweights[h * NB + col];

  float part = 0.0f;
#pragma unroll
  for (int r = 0; r < 8; ++r) {  // all 8 D elements of a lane share column N
    float d = co[r] - cp[r];
    part += d * d;
  }
  part *= bw * wt;

  spart[tid] = part;
  __syncthreads();

  // Fixed-order sequential sum -> deterministic loss.
  if (tid == 0) {
    float s = 0.0f;
    for (int i = 0; i < 512; ++i) s += spart[i];
    out[0] = s / (float)NBATCH;
  }
}

// ---------------------------------------------------------------------------
extern "C" void kernel_launch(void* const* d_in, const int* in_sizes, int n_in,
                              void* d_out, int out_size, void* d_ws,
                              size_t ws_size, hipStream_t stream) {
  const float* obs = (const float*)d_in[0];
  const float* pred = (const float*)d_in[1];
  const unsigned char* mask = (const unsigned char*)d_in[2]; // numpy bool = 1B
  const int* hidx = (const int*)d_in[3];
  const float* edges = (const float*)d_in[4];
  const float* mids = (const float*)d_in[5];
  const float* bweights = (const float*)d_in[6];

  unsigned int* cnt = (unsigned int*)d_ws; // 2*64*64 u32 = 32KB
  float* out = (float*)d_out;

  zero_counts_kernel<<<(2 * NBATCH * NB + 255) / 256, 256, 0, stream>>>(cnt);
  hist_kernel<<<NBATCH * BPB, HT, 0, stream>>>(obs, pred, mask, edges, cnt);
  finalize_kernel<<<1, 512, 0, stream>>>(cnt, hidx, mids, bweights, out);
}